// DEM_15917148799599
// MI455X (gfx1250) — compile-verified
//
#include <hip/hip_runtime.h>
#include <hip/hip_bf16.h>
#include <math.h>

typedef __attribute__((ext_vector_type(2))) float v2f;
typedef __attribute__((ext_vector_type(8))) float v8f;

#define DIM   4096
#define NK1   64
#define NK2   64
#define TILES 4      // 16-row tiles per block -> 64 rows/block
#define GPAD  66     // padded row stride for g in LDS (bank-conflict avoidance)

__device__ __forceinline__ float sigmoid_stable(float z) {
  if (z >= 0.f) { float e = __expf(-z); return 1.f / (1.f + e); }
  float e = __expf(z); return e / (1.f + e);
}
__device__ __forceinline__ float softplus_stable(float u) {
  return fmaxf(u, 0.f) + log1pf(__expf(-fabsf(u)));
}

__global__ __launch_bounds__(256) void dem_fused_kernel(
    const float* __restrict__ x, const float* __restrict__ Vm,
    const float* __restrict__ Wm, const float* __restrict__ cv,
    const float* __restrict__ bv, float* __restrict__ out) {
  __shared__ __align__(32) float s_part[TILES][4][32 * 8]; // partial accumulators
  __shared__ float s_bx[TILES][32];
  __shared__ float s_xx[TILES][32];
  __shared__ float s_g [TILES][16 * GPAD];                 // sigmoid(g) tile
  __shared__ float s_rs[TILES][16];                        // per-row softplus sums

  const int lane  = threadIdx.x & 31;
  const int wave  = threadIdx.x >> 5;
  const int tile  = wave >> 1;     // 4 row-tiles per block
  const int khalf = wave & 1;      // K-split: each wave covers half of DIM
  const int hi    = lane >> 4;     // lane half (A/B frag layout)
  const int l16   = lane & 15;
  const int kb    = hi << 1;       // K offset within 4-wide step: {0,2}

  const int rowBase = blockIdx.x * (TILES * 16) + tile * 16;
  const float* xrow = x  + (size_t)(rowBase + l16) * DIM;
  const float* v0   = Vm + (size_t)(l16 +  0) * DIM;
  const float* v1   = Vm + (size_t)(l16 + 16) * DIM;
  const float* v2   = Vm + (size_t)(l16 + 32) * DIM;
  const float* v3   = Vm + (size_t)(l16 + 48) * DIM;

  v8f a0 = {}, a1 = {}, a2 = {}, a3 = {};
  float bx = 0.f, xx = 0.f;

  const int dBeg = khalf * (DIM / 2);
  const int dEnd = dBeg + (DIM / 2);
  for (int d0 = dBeg; d0 < dEnd; d0 += 32) {
    __builtin_prefetch(xrow + d0 + 1024, 0, 0);   // global_prefetch_b8, x stream
#pragma unroll
    for (int dd = 0; dd < 32; dd += 4) {
      const int d = d0 + dd + kb;
      v2f av  = *(const v2f*)(xrow + d);
      v2f b0v = *(const v2f*)(v0 + d);
      v2f b1v = *(const v2f*)(v1 + d);
      v2f b2v = *(const v2f*)(v2 + d);
      v2f b3v = *(const v2f*)(v3 + d);
      v2f bbv = *(const v2f*)(bv + d);
      a0 = __builtin_amdgcn_wmma_f32_16x16x4_f32(false, av, false, b0v, (short)0, a0, false, false);
      a1 = __builtin_amdgcn_wmma_f32_16x16x4_f32(false, av, false, b1v, (short)0, a1, false, false);
      a2 = __builtin_amdgcn_wmma_f32_16x16x4_f32(false, av, false, b2v, (short)0, a2, false, false);
      a3 = __builtin_amdgcn_wmma_f32_16x16x4_f32(false, av, false, b3v, (short)0, a3, false, false);
      bx = fmaf(av.x, bbv.x, fmaf(av.y, bbv.y, bx));   // b . x  (fused)
      xx = fmaf(av.x, av.x,  fmaf(av.y, av.y,  xx));   // ||x||^2 (fused)
    }
  }

  // merge K-split partials through LDS (wave 2t+1 -> wave 2t)
  if (khalf == 1) {
    *(v8f*)&s_part[tile][0][lane * 8] = a0;
    *(v8f*)&s_part[tile][1][lane * 8] = a1;
    *(v8f*)&s_part[tile][2][lane * 8] = a2;
    *(v8f*)&s_part[tile][3][lane * 8] = a3;
    s_bx[tile][lane] = bx;
    s_xx[tile][lane] = xx;
  }
  __syncthreads();
  if (khalf == 1) return;

  a0 += *(const v8f*)&s_part[tile][0][lane * 8];
  a1 += *(const v8f*)&s_part[tile][1][lane * 8];
  a2 += *(const v8f*)&s_part[tile][2][lane * 8];
  a3 += *(const v8f*)&s_part[tile][3][lane * 8];
  bx += s_bx[tile][lane];
  xx += s_xx[tile][lane];
  bx += __shfl_xor(bx, 16, 32);   // combine the two lane-half K subsets per row
  xx += __shfl_xor(xx, 16, 32);

  // g = sigmoid(x V^T) -> LDS (C/D layout: VGPR i holds rows i and i+8)
  float* g = &s_g[tile][0];
#pragma unroll
  for (int i = 0; i < 8; ++i) {
    const int r = i + 8 * hi;
    g[r * GPAD +  0 + l16] = sigmoid_stable(a0[i]);
    g[r * GPAD + 16 + l16] = sigmoid_stable(a1[i]);
    g[r * GPAD + 32 + l16] = sigmoid_stable(a2[i]);
    g[r * GPAD + 48 + l16] = sigmoid_stable(a3[i]);
  }

  // u = g W^T via WMMA (A frags from LDS, B frags from W; W is L2-hot)
  v8f u0 = {}, u1 = {}, u2 = {}, u3 = {};
  const float* w0 = Wm + (l16 +  0) * NK1;
  const float* w1 = Wm + (l16 + 16) * NK1;
  const float* w2 = Wm + (l16 + 32) * NK1;
  const float* w3 = Wm + (l16 + 48) * NK1;
#pragma unroll
  for (int k = 0; k < NK1; k += 4) {
    v2f ag  = *(const v2f*)&g[l16 * GPAD + k + kb];
    v2f q0v = *(const v2f*)(w0 + k + kb);
    v2f q1v = *(const v2f*)(w1 + k + kb);
    v2f q2v = *(const v2f*)(w2 + k + kb);
    v2f q3v = *(const v2f*)(w3 + k + kb);
    u0 = __builtin_amdgcn_wmma_f32_16x16x4_f32(false, ag, false, q0v, (short)0, u0, false, false);
    u1 = __builtin_amdgcn_wmma_f32_16x16x4_f32(false, ag, false, q1v, (short)0, u1, false, false);
    u2 = __builtin_amdgcn_wmma_f32_16x16x4_f32(false, ag, false, q2v, (short)0, u2, false, false);
    u3 = __builtin_amdgcn_wmma_f32_16x16x4_f32(false, ag, false, q3v, (short)0, u3, false, false);
  }

  const float c0 = cv[l16], c1 = cv[l16 + 16], c2 = cv[l16 + 32], c3 = cv[l16 + 48];
#pragma unroll
  for (int i = 0; i < 8; ++i) {
    float s = softplus_stable(u0[i] + c0) + softplus_stable(u1[i] + c1)
            + softplus_stable(u2[i] + c2) + softplus_stable(u3[i] + c3);
    s += __shfl_xor(s, 1, 32);
    s += __shfl_xor(s, 2, 32);
    s += __shfl_xor(s, 4, 32);
    s += __shfl_xor(s, 8, 32);
    if (l16 == 0) s_rs[tile][i + 8 * hi] = s;   // lane0 -> row i, lane16 -> row i+8
  }
  if (hi == 0) {
    float y = s_rs[tile][l16] + bx - 0.5f * xx;
    out[rowBase + l16] = __expf(y);
  }
}

extern "C" void kernel_launch(void* const* d_in, const int* in_sizes, int n_in,
                              void* d_out, int out_size, void* d_ws, size_t ws_size,
                              hipStream_t stream) {
  const float* x  = (const float*)d_in[0];
  const float* Vm = (const float*)d_in[1];
  const float* Wm = (const float*)d_in[2];
  const float* cv = (const float*)d_in[3];
  const float* bv = (const float*)d_in[4];
  float* out = (float*)d_out;
  const int B = in_sizes[0] / DIM;          // 16384
  const int grid = B / (TILES * 16);        // 64 rows per block -> 256 blocks
  dem_fused_kernel<<<grid, 256, 0, stream>>>(x, Vm, Wm, cv, bv, out);
}